// LinearAttention_62895501082635
// MI455X (gfx1250) — compile-verified
//
#include <hip/hip_runtime.h>
#include <math.h>

typedef __attribute__((ext_vector_type(2))) float v2f;
typedef __attribute__((ext_vector_type(8))) float v8f;

#define B_   16
#define CDIM 512
#define F_   4096
#define H_   8
#define OC3  1536
#define EPS_ 1e-5f
#define NEG_INF_ (-3.402823466e38f)
#define NSPLIT 8   // n-split for context GEMM

__device__ __forceinline__ v8f v8f_zero() {
  v8f z;
#pragma unroll
  for (int i = 0; i < 8; ++i) z[i] = 0.f;
  return z;
}

__device__ __forceinline__ v8f wmma_f32(v2f a, v2f b, v8f c) {
  // D = A(16x4,f32) * B(4x16,f32) + C(16x16,f32)
  return __builtin_amdgcn_wmma_f32_16x16x4_f32(
      /*neg_a=*/false, a, /*neg_b=*/false, b,
      /*c_mod=*/(short)0, c, /*reuse_a=*/false, /*reuse_b=*/false);
}

// low 32 bits of a flat address in the LDS aperture = wave-relative LDS offset
__device__ __forceinline__ unsigned lds_off32(const void* p) {
  return (unsigned)(unsigned long long)p;
}

// async global -> LDS, 16 bytes per lane (tracked with ASYNCcnt)
__device__ __forceinline__ void async_b128(unsigned lds_addr, const void* gptr) {
  unsigned long long ga = (unsigned long long)gptr;
  asm volatile("global_load_async_to_lds_b128 %0, %1, off"
               :: "v"(lds_addr), "v"(ga) : "memory");
}

// ---------------------------------------------------------------------------
// Kernel 1: qkv[b, o, f] = sum_c W_qkv[o, c] * x[b, c, f]
// grid = (F/128, OC3/64, B), block = 256 (8 waves).
// Block tile 64x128: wave -> 4 M-tiles sharing one N=16 strip.
// x K-chunks (32x128) staged in LDS via async b128, double buffered.
// ---------------------------------------------------------------------------
#define KC1 32
#define LXS 144   // padded row stride (floats): 2 rows apart -> +32 banks
__global__ __launch_bounds__(256)
void k_qkv_gemm(const float* __restrict__ x, const float* __restrict__ w,
                float* __restrict__ qkv) {
  __shared__ float lx[2][KC1 * LXS];

  const int tid  = threadIdx.x;
  const int lane = tid & 31;
  const int wave = tid >> 5;
  const int nblk = blockIdx.x * 128;
  const int m0   = blockIdx.y * 64;
  const int b    = blockIdx.z;
  const int ln   = lane & 15;
  const int koff = (lane >> 4) << 1;

  const float* xb = x + ((size_t)b * CDIM) * F_ + nblk;
  // this thread's 4 async elements per chunk: idx = u*256+tid -> row kk, 4-float group j4
  int arow[4], acol[4];
  unsigned aldst[4];
#pragma unroll
  for (int u = 0; u < 4; ++u) {
    const int idx = u * 256 + tid;
    arow[u] = idx >> 5;          // 0..31
    acol[u] = (idx & 31) * 4;    // 0..124
    aldst[u] = 0;                // filled per buffer below
  }

  v8f acc[4];
#pragma unroll
  for (int t = 0; t < 4; ++t) acc[t] = v8f_zero();

  // issue chunk 0 into buffer 0
#pragma unroll
  for (int u = 0; u < 4; ++u)
    async_b128(lds_off32(&lx[0][arow[u] * LXS + acol[u]]),
               xb + (size_t)arow[u] * F_ + acol[u]);

  for (int kc = 0; kc < CDIM; kc += KC1) {
    const int buf = (kc / KC1) & 1;
    if (kc + KC1 < CDIM) {
      const int nbuf = buf ^ 1;
#pragma unroll
      for (int u = 0; u < 4; ++u)
        async_b128(lds_off32(&lx[nbuf][arow[u] * LXS + acol[u]]),
                   xb + (size_t)(kc + KC1 + arow[u]) * F_ + acol[u]);
      asm volatile("s_wait_asynccnt 0x4" ::: "memory");
    } else {
      asm volatile("s_wait_asynccnt 0x0" ::: "memory");
    }
    __syncthreads();

    const float* lxb = &lx[buf][0];
#pragma unroll
    for (int kk = 0; kk < KC1; kk += 4) {
      v2f bb;
      bb.x = lxb[(kk + koff) * LXS + wave * 16 + ln];
      bb.y = lxb[(kk + koff + 1) * LXS + wave * 16 + ln];
#pragma unroll
      for (int t = 0; t < 4; ++t) {
        v2f a = *(const v2f*)(w + (size_t)(m0 + t * 16 + ln) * CDIM + kc + kk + koff);
        acc[t] = wmma_f32(a, bb, acc[t]);
      }
    }
    __syncthreads();
  }

  const int half = (lane >> 4) * 8;
#pragma unroll
  for (int t = 0; t < 4; ++t) {
    float* op = qkv + ((size_t)b * OC3 + m0 + t * 16 + half) * F_ + nblk + wave * 16 + ln;
#pragma unroll
    for (int r = 0; r < 8; ++r) op[(size_t)r * F_] = acc[t][r];
  }
}

// ---------------------------------------------------------------------------
// Kernel 2: softmax of k over n (row-contiguous), in place.
// ---------------------------------------------------------------------------
__global__ __launch_bounds__(256)
void k_softmax_k(float* __restrict__ qkv) {
  const int row = blockIdx.x;            // 0 .. B*512-1
  const int b = row >> 9;
  const int c = row & 511;
  float* p = qkv + ((size_t)b * OC3 + 512 + c) * F_;

  const int tid = threadIdx.x;
  const int wave = tid >> 5;

  float v[16];
  float mx = NEG_INF_;
#pragma unroll
  for (int i = 0; i < 16; ++i) {
    v[i] = p[tid + i * 256];
    mx = fmaxf(mx, v[i]);
  }
#pragma unroll
  for (int o = 16; o > 0; o >>= 1) mx = fmaxf(mx, __shfl_xor(mx, o));

  __shared__ float sred[8];
  __shared__ float bc[2];
  if ((tid & 31) == 0) sred[wave] = mx;
  __syncthreads();
  if (tid == 0) {
    float m = sred[0];
#pragma unroll
    for (int i = 1; i < 8; ++i) m = fmaxf(m, sred[i]);
    bc[0] = m;
  }
  __syncthreads();
  mx = bc[0];

  float s = 0.f;
#pragma unroll
  for (int i = 0; i < 16; ++i) {
    v[i] = __expf(v[i] - mx);
    s += v[i];
  }
#pragma unroll
  for (int o = 16; o > 0; o >>= 1) s += __shfl_xor(s, o);
  __syncthreads();
  if ((tid & 31) == 0) sred[wave] = s;
  __syncthreads();
  if (tid == 0) {
    float t = 0.f;
#pragma unroll
    for (int i = 0; i < 8; ++i) t += sred[i];
    bc[1] = t;
  }
  __syncthreads();
  const float inv = 1.f / bc[1];
#pragma unroll
  for (int i = 0; i < 16; ++i) p[tid + i * 256] = v[i] * inv;
}

// ---------------------------------------------------------------------------
// Kernel 3: softmax of q over d (64, stride F), * scale, in place.
// ---------------------------------------------------------------------------
__global__ __launch_bounds__(256)
void k_softmax_q(float* __restrict__ qkv) {
  const int idx = blockIdx.x * 256 + threadIdx.x;   // 0 .. B*H*F-1
  const int n  = idx & (F_ - 1);
  const int bh = idx >> 12;
  const int b  = bh >> 3;
  const int h  = bh & 7;
  float* p = qkv + ((size_t)b * OC3 + h * 64) * F_ + n;

  float v[64];
  float mx = NEG_INF_;
#pragma unroll
  for (int d = 0; d < 64; ++d) {
    v[d] = p[(size_t)d * F_];
    mx = fmaxf(mx, v[d]);
  }
  float s = 0.f;
#pragma unroll
  for (int d = 0; d < 64; ++d) {
    v[d] = __expf(v[d] - mx);
    s += v[d];
  }
  const float inv = 0.125f / s;   // scale = 1/sqrt(64) folded in
#pragma unroll
  for (int d = 0; d < 64; ++d) p[(size_t)d * F_] = v[d] * inv;
}

// ---------------------------------------------------------------------------
// Kernel 4a: partial ctx over an n-chunk of F/NSPLIT.
// grid = (NSPLIT, B*H); wave -> two adjacent 16x16 tiles sharing A rows.
// ---------------------------------------------------------------------------
__global__ __launch_bounds__(256)
void k_context_part(const float* __restrict__ qkv, float* __restrict__ ctxp) {
  const int lane = threadIdx.x & 31;
  const int wave = threadIdx.x >> 5;
  const int s  = blockIdx.x;             // n-chunk
  const int bh = blockIdx.y;
  const int b = bh >> 3, h = bh & 7;
  const int ln = lane & 15;
  const int koff = (lane >> 4) << 1;
  const int t0 = wave * 2;
  const int m0 = (t0 >> 2) << 4;     // d tile base
  const int n0 = (t0 & 3) << 4;      // e tile base (and +16)
  const int kbeg = s * (F_ / NSPLIT);

  const float* kb  = qkv + ((size_t)b * OC3 + 512  + h * 64 + m0 + ln) * F_ + kbeg + koff;
  const float* vb0 = qkv + ((size_t)b * OC3 + 1024 + h * 64 + n0 + ln) * F_ + kbeg + koff;
  const float* vb1 = vb0 + (size_t)16 * F_;

  v8f acc0 = v8f_zero(), acc1 = v8f_zero();
  for (int k = 0; k < F_ / NSPLIT; k += 4) {
    v2f a  = *(const v2f*)(kb + k);
    v2f b0 = *(const v2f*)(vb0 + k);
    v2f b1 = *(const v2f*)(vb1 + k);
    acc0 = wmma_f32(a, b0, acc0);
    acc1 = wmma_f32(a, b1, acc1);
  }

  const int half = (lane >> 4) * 8;
  float* o0 = ctxp + (((size_t)s * (B_ * H_) + bh) * 64 + m0 + half) * 64 + n0 + ln;
#pragma unroll
  for (int r = 0; r < 8; ++r) {
    o0[r * 64]      = acc0[r];
    o0[r * 64 + 16] = acc1[r];
  }
}

// Kernel 4b: ctx = sum over NSPLIT partials.
__global__ __launch_bounds__(256)
void k_context_reduce(const float* __restrict__ ctxp, float* __restrict__ ctx) {
  const size_t i = (size_t)blockIdx.x * 256 + threadIdx.x;  // < B*H*64*64
  const size_t stride = (size_t)B_ * H_ * 64 * 64;
  float s = 0.f;
#pragma unroll
  for (int c = 0; c < NSPLIT; ++c) s += ctxp[c * stride + i];
  ctx[i] = s;
}

// ---------------------------------------------------------------------------
// Kernel 5: attn[bh][e][n] = sum_d ctx[d,e] * q~[d,n]; writes into dead k-slot.
// ---------------------------------------------------------------------------
__global__ __launch_bounds__(256)
void k_attn(float* __restrict__ qkv, const float* __restrict__ ctx) {
  const int lane = threadIdx.x & 31;
  const int wave = threadIdx.x >> 5;
  const int bh = blockIdx.y;
  const int b = bh >> 3, h = bh & 7;
  const int n0 = blockIdx.x * 128 + wave * 16;
  const int ln = lane & 15;
  const int koff = (lane >> 4) << 1;

  const float* qb = qkv + ((size_t)b * OC3 + h * 64) * F_ + n0 + ln;       // q~[d,n]
  const float* cb = ctx + (size_t)bh * 64 * 64;                             // ctx[d,e]
  float* ab = qkv + ((size_t)b * OC3 + 512 + h * 64) * F_ + n0 + ln;        // out e rows

  v8f acc[4];
#pragma unroll
  for (int t = 0; t < 4; ++t) acc[t] = v8f_zero();

  for (int k = 0; k < 64; k += 4) {
    v2f bb;
    bb.x = qb[(size_t)(k + koff) * F_];
    bb.y = qb[(size_t)(k + koff + 1) * F_];
#pragma unroll
    for (int t = 0; t < 4; ++t) {
      const int row = t * 16 + ln;               // e
      v2f a;
      a.x = cb[(k + koff) * 64 + row];
      a.y = cb[(k + koff + 1) * 64 + row];
      acc[t] = wmma_f32(a, bb, acc[t]);
    }
  }

  const int half = (lane >> 4) * 8;
#pragma unroll
  for (int t = 0; t < 4; ++t) {
    float* op = ab + (size_t)(t * 16 + half) * F_;
#pragma unroll
    for (int r = 0; r < 8; ++r) op[(size_t)r * F_] = acc[t][r];
  }
}

// ---------------------------------------------------------------------------
// Kernel 6: out = LN_channel(W_out @ attn + b_out) * g, fused.
// grid = (F/16, B); block owns the full 512-channel column strip of width 16.
// attn tiles staged in LDS via async b128; cross-wave mean/var reduction.
// ---------------------------------------------------------------------------
__global__ __launch_bounds__(256)
void k_outln(const float* __restrict__ qkv, const float* __restrict__ wout,
             const float* __restrict__ bout, const float* __restrict__ g,
             float* __restrict__ out) {
  __shared__ float lb[64 * 16];
  __shared__ float red1[8][16], red2[8][16];
  __shared__ float smean[16], srstd[16];

  const int tid  = threadIdx.x;
  const int lane = tid & 31;
  const int wave = tid >> 5;
  const int n0 = blockIdx.x * 16;
  const int b  = blockIdx.y;
  const int ln = lane & 15;
  const int koff = (lane >> 4) << 1;

  const float* attnb = qkv + ((size_t)b * OC3 + 512) * F_;   // attn [c][f]
  const int arow = tid >> 2;          // 0..63
  const int acol = (tid & 3) * 4;     // 0,4,8,12
  const unsigned aldst = lds_off32(&lb[arow * 16 + acol]);

  v8f acc[4];
#pragma unroll
  for (int t = 0; t < 4; ++t) acc[t] = v8f_zero();

  for (int kc = 0; kc < CDIM; kc += 64) {
    async_b128(aldst, attnb + (size_t)(kc + arow) * F_ + n0 + acol);
    asm volatile("s_wait_asynccnt 0x0" ::: "memory");
    __syncthreads();
#pragma unroll
    for (int kk = 0; kk < 64; kk += 4) {
      v2f bb;
      bb.x = lb[(kk + koff) * 16 + ln];
      bb.y = lb[(kk + koff + 1) * 16 + ln];
      const int k = kc + kk;
#pragma unroll
      for (int t = 0; t < 4; ++t) {
        const float* wr = wout + (size_t)(wave * 64 + t * 16 + ln) * CDIM + koff + k;
        v2f a = *(const v2f*)wr;
        acc[t] = wmma_f32(a, bb, acc[t]);
      }
    }
    __syncthreads();
  }

  // bias + per-column stats
  const int half = (lane >> 4) * 8;
  float s1 = 0.f, s2 = 0.f;
#pragma unroll
  for (int t = 0; t < 4; ++t) {
#pragma unroll
    for (int r = 0; r < 8; ++r) {
      const int row = wave * 64 + t * 16 + half + r;
      float v = acc[t][r] + bout[row];
      acc[t][r] = v;
      s1 += v;
      s2 += v * v;
    }
  }
  s1 += __shfl_xor(s1, 16);
  s2 += __shfl_xor(s2, 16);
  if (lane < 16) { red1[wave][lane] = s1; red2[wave][lane] = s2; }
  __syncthreads();
  if (tid < 16) {
    float m1 = 0.f, m2 = 0.f;
#pragma unroll
    for (int w2 = 0; w2 < 8; ++w2) { m1 += red1[w2][tid]; m2 += red2[w2][tid]; }
    const float mean = m1 * (1.f / 512.f);
    const float var  = m2 * (1.f / 512.f) - mean * mean;
    smean[tid] = mean;
    srstd[tid] = rsqrtf(var + EPS_);
  }
  __syncthreads();
  const float mean = smean[ln];
  const float rstd = srstd[ln];
#pragma unroll
  for (int t = 0; t < 4; ++t) {
#pragma unroll
    for (int r = 0; r < 8; ++r) {
      const int row = wave * 64 + t * 16 + half + r;
      out[((size_t)b * CDIM + row) * F_ + n0 + ln] =
          (acc[t][r] - mean) * rstd * g[row];
    }
  }
}

// ---------------------------------------------------------------------------
extern "C" void kernel_launch(void* const* d_in, const int* in_sizes, int n_in,
                              void* d_out, int out_size, void* d_ws, size_t ws_size,
                              hipStream_t stream) {
  (void)in_sizes; (void)n_in; (void)out_size; (void)ws_size;
  const float* x    = (const float*)d_in[0];
  const float* wqkv = (const float*)d_in[1];
  const float* wout = (const float*)d_in[2];
  const float* bout = (const float*)d_in[3];
  const float* g    = (const float*)d_in[4];
  float* out = (float*)d_out;

  float* qkv  = (float*)d_ws;                              // [B][1536][F] fp32
  float* ctxp = qkv + (size_t)B_ * OC3 * F_;               // [NSPLIT][B*H][64][64]
  float* ctx  = ctxp + (size_t)NSPLIT * B_ * H_ * 64 * 64; // [B*H][64][64]

  k_qkv_gemm<<<dim3(F_ / 128, OC3 / 64, B_), 256, 0, stream>>>(x, wqkv, qkv);
  k_softmax_k<<<dim3(B_ * 512), 256, 0, stream>>>(qkv);
  k_softmax_q<<<dim3(B_ * H_ * F_ / 256), 256, 0, stream>>>(qkv);
  k_context_part<<<dim3(NSPLIT, B_ * H_), 256, 0, stream>>>(qkv, ctxp);
  k_context_reduce<<<dim3(B_ * H_ * 64 * 64 / 256), 256, 0, stream>>>(ctxp, ctx);
  k_attn<<<dim3(F_ / 128, B_ * H_), 256, 0, stream>>>(qkv, ctx);
  k_outln<<<dim3(F_ / 16, B_), 256, 0, stream>>>(qkv, wout, bout, g, out);
}